// Gridnet_60619168416447
// MI455X (gfx1250) — compile-verified
//
#include <hip/hip_runtime.h>

// GridNet step for MI455X (gfx1250, wave32).
// One WG = one 8^3 spatial block x 4 batches; 512 threads = 16 wave32s, one
// thread per inner cell. 27 per-position tap weights live in VGPRs for the WG
// lifetime. The 10^3 halo block is staged into LDS with CDNA5 async
// global->LDS loads. The cross-wave mean/var reduction is finished with a
// single V_WMMA_F32_16X16X4_F32 (B=ones): one WMMA broadcasts BOTH the sum
// and sum-of-squares totals to every lane, replacing 32 LDS reads + 30 adds.
// Sigmoid uses v_exp_f32 + v_rcp_f32 (no IEEE divide expansion).

#define BS     8
#define H      10          // BS + 2 (halo)
#define HVOL   1000        // 10^3
#define CELLS  512         // 8^3 threads, one per inner cell
#define DIM    64
#define VOL    (DIM*DIM*DIM)
#define BATCH  16
#define BPG    4           // batches per workgroup
#define ITERS  8
#define EPS    1e-5f

#if defined(__has_builtin)
#  if __has_builtin(__builtin_amdgcn_global_load_async_to_lds_b32)
#    define HAVE_ASYNC_LDS 1
#  endif
#  if __has_builtin(__builtin_amdgcn_s_wait_asynccnt)
#    define ASYNC_WAIT0() __builtin_amdgcn_s_wait_asynccnt(0)
#  else
#    define ASYNC_WAIT0() asm volatile("s_wait_asynccnt 0" ::: "memory")
#  endif
#  if __has_builtin(__builtin_amdgcn_wmma_f32_16x16x4_f32)
#    define HAVE_WMMA_RED 1
#  endif
#  if __has_builtin(__builtin_amdgcn_rcpf)
#    define FAST_RCP(x) __builtin_amdgcn_rcpf(x)
#  else
#    define FAST_RCP(x) (1.0f / (x))
#  endif
#else
#  define FAST_RCP(x) (1.0f / (x))
#endif

#ifdef HAVE_ASYNC_LDS
// Builtin signature (from clang diagnostic): src is AS(1) int*, dst AS(3) int*.
#  define GPTR(p) ((__attribute__((address_space(1))) int*)(p))
#  define LPTR(p) ((__attribute__((address_space(3))) int*)(p))
#endif

typedef __attribute__((ext_vector_type(2))) float v2f;
typedef __attribute__((ext_vector_type(8))) float v8f;

__global__ __launch_bounds__(CELLS, 2) void gridnet_kernel(
    const float* __restrict__ weight,   // [27][64][64][64]
    const float* __restrict__ bias,     // [64][64][64]
    const float* __restrict__ rscale,   // [64][64][64]
    const float* __restrict__ x,        // [16][64][64][64]
    float* __restrict__ out)            // [16][64][64][64]
{
    __shared__ float blk[HVOL];
    __shared__ float red[32];           // [0..15]=sum partials, [16..31]=sumsq

    const int tid  = threadIdx.x;
    const int lk   = tid & 7, ln = (tid >> 3) & 7, lm = tid >> 6;
    const int sb   = blockIdx.x >> 2;         // spatial block 0..511
    const int bg   = blockIdx.x & 3;          // batch group 0..3
    const int bk   = sb & 7, bn = (sb >> 3) & 7, bm = sb >> 6;
    const int gm   = bm * BS + lm, gn = bn * BS + ln, gk = bk * BS + lk;
    const int cell = gm * (DIM*DIM) + gn * DIM + gk;
    const int cidx = (lm + 1) * 100 + (ln + 1) * 10 + (lk + 1);
    const int wave = tid >> 5;
    const int lane = tid & 31;

#ifdef HAVE_WMMA_RED
    // A-matrix addressing for V_WMMA_F32_16X16X4_F32 (16x4, f32):
    //   lanes 0-15 hold A[m][0],A[m][1]; lanes 16-31 hold A[m][2],A[m][3],
    //   with m = lane & 15.  We want rowsum_m = sum_k red[sel*16 + 4k + r],
    //   sel=(m>>2)&1 (sum vs sumsq), r=m&3.  Loop-invariant index:
    const int mrow  = lane & 15;
    const int khalf = lane >> 4;
    const int aidx  = ((mrow >> 2) & 1) * 16 + khalf * 8 + (mrow & 3);
    const v2f bones = {1.0f, 1.0f};
#endif

    // --- per-thread parameter cache (reused across BPG batches x 8 iters) ---
    float w[27];
    float wsum = 0.f;
    #pragma unroll
    for (int t = 0; t < 27; ++t) {
        w[t] = weight[t * VOL + cell];
        wsum += w[t];
    }
    const float b0 = bias[cell];
    const float rs = rscale[cell];

    for (int bb = 0; bb < BPG; ++bb) {
        const int b = bg * BPG + bb;
        const float* xb = x + b * VOL;

        // --- stage 10^3 block into LDS (zero halo at volume boundary) ---
        for (int e = tid; e < HVOL; e += CELLS) {
            const int m = e / 100, r = e - m * 100;
            const int n = r / 10,  k = r - n * 10;
            const int sm = bm * BS + m - 1;
            const int sn = bn * BS + n - 1;
            const int sk = bk * BS + k - 1;
            const bool in = ((unsigned)sm < (unsigned)DIM) &
                            ((unsigned)sn < (unsigned)DIM) &
                            ((unsigned)sk < (unsigned)DIM);
#ifdef HAVE_ASYNC_LDS
            if (in) {
                __builtin_amdgcn_global_load_async_to_lds_b32(
                    GPTR(xb + sm * (DIM*DIM) + sn * DIM + sk),
                    LPTR(&blk[e]), 0, 0);
            } else {
                blk[e] = 0.f;
            }
#else
            blk[e] = in ? xb[sm * (DIM*DIM) + sn * DIM + sk] : 0.f;
#endif
        }
        // prefetch next batch's cell region into cache while we compute
        if (bb + 1 < BPG) __builtin_prefetch(xb + VOL + cell, 0, 1);
#ifdef HAVE_ASYNC_LDS
        ASYNC_WAIT0();
#endif
        __syncthreads();

        for (int it = 0; it < ITERS; ++it) {
            // ---- mean / var over the full 10^3 region ----
            float v0 = blk[tid];
            float s = v0, q = v0 * v0;
            if (tid < HVOL - CELLS) {
                float v1 = blk[tid + CELLS];
                s += v1; q += v1 * v1;
            }
            #pragma unroll
            for (int off = 16; off > 0; off >>= 1) {
                s += __shfl_down(s, off, 32);
                q += __shfl_down(q, off, 32);
            }
            if (lane == 0) { red[wave] = s; red[16 + wave] = q; }
            __syncthreads();

            float S, Q;
#ifdef HAVE_WMMA_RED
            {
                // One 16x16x4 f32 WMMA sums all 16 partials of BOTH
                // quantities and broadcasts the results to every lane.
                v2f a;
                a.x = red[aidx];
                a.y = red[aidx + 4];
                v8f c = {};
                v8f d = __builtin_amdgcn_wmma_f32_16x16x4_f32(
                    false, a, false, bones, (short)0, c, false, false);
                S = (d[0] + d[1]) + (d[2] + d[3]);
                Q = (d[4] + d[5]) + (d[6] + d[7]);
            }
#else
            S = 0.f; Q = 0.f;
            #pragma unroll
            for (int i = 0; i < 16; ++i) { S += red[i]; Q += red[16 + i]; }
#endif
            const float mean = S * (1.0f / HVOL);
            const float var  = Q * (1.0f / HVOL) - mean * mean;
            const float rsig = __frsqrt_rn(var + EPS);

            // ---- 27-tap stencil on RAW values; fold layernorm in after:
            //      sum_t w_t*(v-mean)*rsig == rsig*(sum_t w_t*v - mean*wsum)
            float acc = 0.f;
            const float* c = &blk[cidx];
            #pragma unroll
            for (int dx = 0; dx < 3; ++dx)
                #pragma unroll
                for (int dy = 0; dy < 3; ++dy)
                    #pragma unroll
                    for (int dz = 0; dz < 3; ++dz)
                        acc = fmaf(w[dx * 9 + dy * 3 + dz],
                                   c[(dx - 1) * 100 + (dy - 1) * 10 + (dz - 1)],
                                   acc);

            const float dot = rsig * (acc - mean * wsum);
            const float z   = dot + b0;
            const float sig = FAST_RCP(1.0f + __expf(-z));   // v_exp + v_rcp
            const float cur = blk[cidx];
            const float nv  = cur + rs * (z * sig);

            __syncthreads();          // all neighbor reads done
            blk[cidx] = nv;
            __syncthreads();          // update visible for next reduction
        }

        out[b * VOL + cell] = blk[cidx];
        __syncthreads();              // keep blk stable until everyone stored
    }
}

extern "C" void kernel_launch(void* const* d_in, const int* in_sizes, int n_in,
                              void* d_out, int out_size, void* d_ws, size_t ws_size,
                              hipStream_t stream) {
    const float* weight = (const float*)d_in[0];
    const float* bias   = (const float*)d_in[1];
    const float* rscale = (const float*)d_in[2];
    const float* x      = (const float*)d_in[3];
    float* out          = (float*)d_out;

    dim3 grid(512 * (BATCH / BPG));   // 2048 workgroups
    dim3 block(CELLS);                // 512 threads = 16 wave32s
    gridnet_kernel<<<grid, block, 0, stream>>>(weight, bias, rscale, x, out);
}